// ModelPassMessage_76192719831680
// MI455X (gfx1250) — compile-verified
//
#include <hip/hip_runtime.h>

// GNN message passing (copy_u/copy_e + segment_sum, 7 conv layers) for MI455X.
// Bottleneck: L2-resident gather/scatter-add (~360M f32 atomics; whole working
// set of 38MB << 192MB L2). Dense [50000,129]x[129,64] GEMMs (5.8 GFLOP total)
// run on fp32 WMMA (v_wmma_f32_16x16x4_f32) at full precision. Weights are
// pre-packed K-major/pair-interleaved/zero-padded so the WMMA K-loop is
// branch-free with coalesced b64 fragment loads.

#define NNODES 50000
#define NEDGES 800000
#define FEAT   64
#define KDIM   129   // 2*64 + 1
#define NLAYER 7

static_assert(NNODES % 16 == 0, "row tiles must divide evenly (no predication)");

typedef __attribute__((ext_vector_type(2))) float v2f;
typedef __attribute__((ext_vector_type(8))) float v8f;

__global__ __launch_bounds__(256) void zero_f32(float* __restrict__ p, int n) {
  int i = blockIdx.x * blockDim.x + threadIdx.x;
  if (i < n) p[i] = 0.f;
}

// he_aggr[dst[e]] += edge_feat[e]
__global__ __launch_bounds__(256) void he_aggr_kernel(const float* __restrict__ ef,
                                                      const int* __restrict__ dst,
                                                      float* __restrict__ he, int n_edges) {
  int e = blockIdx.x * blockDim.x + threadIdx.x;
  if (e < n_edges) atomicAdd(&he[dst[e]], ef[e]);
}

// aggr[dst[e]][:] += h[src[e]][:]  -- 16 threads per edge, float4 per thread.
__global__ __launch_bounds__(256) void scatter_kernel(const float* __restrict__ h,
                                                      const int* __restrict__ src,
                                                      const int* __restrict__ dst,
                                                      float* __restrict__ aggr, int n_edges) {
  int t = blockIdx.x * blockDim.x + threadIdx.x;
  int e = t >> 4;
  if (e >= n_edges) return;
  int f = (t & 15) << 2;
  int s = src[e];
  int d = dst[e];
  float4 v = *(const float4*)(h + (size_t)s * FEAT + f);
  float* o = aggr + (size_t)d * FEAT + f;
  atomicAdd(o + 0, v.x);
  atomicAdd(o + 1, v.y);
  atomicAdd(o + 2, v.z);
  atomicAdd(o + 3, v.w);
}

// Pack all 7 layers' weights into K-major pair-interleaved form:
//   wt2[l][kp][n] = { W_l[n][2*kp], W_l[n][2*kp+1] }   kp in [0,64), n in [0,64)
// zero-padded for n >= n_out (final layer has n_out=2). 7*64*64 float2 = 224KB.
__global__ __launch_bounds__(256) void pack_weights(const float* __restrict__ W1,
                                                    const float* __restrict__ Wmid,
                                                    const float* __restrict__ W2,
                                                    float2* __restrict__ wt2) {
  int i = blockIdx.x * blockDim.x + threadIdx.x;   // l*4096 + kp*64 + n
  if (i >= NLAYER * 64 * 64) return;
  int l  = i >> 12;
  int kp = (i >> 6) & 63;
  int n  = i & 63;
  const float* W;
  int n_out;
  if (l == 0)      { W = W1;                          n_out = 64; }
  else if (l < 6)  { W = Wmid + (size_t)(l - 1) * 64 * KDIM; n_out = 64; }
  else             { W = W2;                          n_out = 2;  }
  float2 v; v.x = 0.f; v.y = 0.f;
  if (n < n_out) {
    v.x = W[(size_t)n * KDIM + 2 * kp];
    v.y = W[(size_t)n * KDIM + 2 * kp + 1];
  }
  wt2[i] = v;
}

// out[m][n] = act( sum_k concat(h,aggr)[m][k]*W[n][k] + he[m]*W[n][128] + b[n] )
// One wave computes a 16-row x (NTILES*16)-col strip via V_WMMA_F32_16X16X4_F32.
// K-loop is branch-free: A is an aligned b64 load from the node row; B is a
// coalesced b64 load from the packed wt2 buffer. ACT: 0=relu, 1=sigmoid, 2=none.
template <int NTILES, int ACT>
__global__ __launch_bounds__(256) void gemm_conv(const float* __restrict__ h,
                                                 const float* __restrict__ aggr,
                                                 const float* __restrict__ he,
                                                 const float2* __restrict__ wt2,
                                                 const float* __restrict__ W,
                                                 const float* __restrict__ b,
                                                 float* __restrict__ out,
                                                 int n_out, int out_stride) {
  const int lane = threadIdx.x & 31;
  const int wave = threadIdx.x >> 5;
  const int row0 = (blockIdx.x * 8 + wave) * 16;
  if (row0 >= NNODES) return;   // wave-uniform: EXEC stays all-ones for WMMA

  const int am = lane & 15;            // A row within tile (M)
  const int kk = (lane >> 4) << 1;     // K sub-offset (0 or 2), same for A and B
  const int row = row0 + am;
  const int bn = lane & 15;            // B column (N)

  v8f acc[NTILES] = {};

  const float* Ab[2] = {h, aggr};
  #pragma unroll
  for (int half = 0; half < 2; ++half) {
    const float* A = Ab[half] + (size_t)row * FEAT + kk;
    #pragma unroll
    for (int k0 = 0; k0 < 64; k0 += 4) {
      v2f a = *(const v2f*)(A + k0);                              // 8B-aligned b64
      const float2* Bk = wt2 + (size_t)(((half * 64 + k0 + kk) >> 1) * 64) + bn;
      #pragma unroll
      for (int t = 0; t < NTILES; ++t) {
        float2 w = Bk[t * 16];                                    // coalesced b64
        v2f bb; bb.x = w.x; bb.y = w.y;
        // (neg_a, A, neg_b, B, c_mod, C, reuse_a, reuse_b)
        acc[t] = __builtin_amdgcn_wmma_f32_16x16x4_f32(false, a, false, bb, (short)0,
                                                       acc[t], false, false);
      }
    }
  }

  // Epilogue: + he[m]*W[n][128] + b[n], activation, store.
  #pragma unroll
  for (int t = 0; t < NTILES; ++t) {
    int nc = t * 16 + (lane & 15);
    if (nc >= n_out) continue;                 // final layer: cols 2..15 dropped
    float wlast = W[(size_t)nc * KDIM + 128];
    float bias  = b[nc];
    #pragma unroll
    for (int r = 0; r < 8; ++r) {
      int orow = row0 + r + ((lane >> 4) << 3);   // C layout: M = r + 8*(lane>=16)
      float v = acc[t][r] + he[orow] * wlast + bias;
      if (ACT == 0)      v = v > 0.f ? v : 0.f;
      else if (ACT == 1) v = 1.f / (1.f + __expf(-v));
      out[(size_t)orow * out_stride + nc] = v;
    }
  }
}

extern "C" void kernel_launch(void* const* d_in, const int* in_sizes, int n_in,
                              void* d_out, int out_size, void* d_ws, size_t ws_size,
                              hipStream_t stream) {
  (void)in_sizes; (void)n_in; (void)out_size; (void)ws_size;
  const float* node_feat = (const float*)d_in[0];
  const float* edge_feat = (const float*)d_in[1];
  const int*   src       = (const int*)d_in[2];
  const int*   dst       = (const int*)d_in[3];
  const float* W1        = (const float*)d_in[4];
  const float* b1        = (const float*)d_in[5];
  const float* Wmid      = (const float*)d_in[6];
  const float* bmid      = (const float*)d_in[7];
  const float* W2        = (const float*)d_in[8];
  const float* b2        = (const float*)d_in[9];
  float* out = (float*)d_out;

  // Workspace layout (all 16B-aligned): he | aggr | hA | hB | wt2
  float*  he   = (float*)d_ws;                       // 50000 floats
  float*  aggr = he + NNODES;                        // 50000*64
  float*  hA   = aggr + (size_t)NNODES * FEAT;       // 50000*64
  float*  hB   = hA   + (size_t)NNODES * FEAT;       // 50000*64
  float2* wt2  = (float2*)(hB + (size_t)NNODES * FEAT);  // 7*64*64 float2

  dim3 blk(256);

  // One-time prep: packed weights + he_aggr = segment_sum(edge_feat, dst).
  pack_weights<<<(NLAYER * 64 * 64 + 255) / 256, blk, 0, stream>>>(W1, Wmid, W2, wt2);
  zero_f32<<<(NNODES + 255) / 256, blk, 0, stream>>>(he, NNODES);
  he_aggr_kernel<<<(NEDGES + 255) / 256, blk, 0, stream>>>(edge_feat, dst, he, NEDGES);

  const int gemm_grid    = (NNODES + 127) / 128;          // 8 waves x 16 rows / block
  const int scatter_grid = ((NEDGES * 16) + 255) / 256;   // 16 threads per edge
  const int zero_grid    = (NNODES * FEAT + 255) / 256;

  const float* hin = node_feat;
  float* bufs[2] = {hA, hB};
  for (int l = 0; l < NLAYER; ++l) {
    zero_f32<<<zero_grid, blk, 0, stream>>>(aggr, NNODES * FEAT);
    scatter_kernel<<<scatter_grid, blk, 0, stream>>>(hin, src, dst, aggr, NEDGES);
    const float2* wt2_l = wt2 + (size_t)l * 64 * 64;
    if (l == 6) {
      // Final layer: N_CLS=2, no activation, write d_out [50000, 2].
      gemm_conv<1, 2><<<gemm_grid, blk, 0, stream>>>(hin, aggr, he, wt2_l, W2, b2,
                                                     out, 2, 2);
    } else {
      const float* W  = (l == 0) ? W1 : Wmid + (size_t)(l - 1) * FEAT * KDIM;
      const float* bb = (l == 0) ? b1 : bmid + (size_t)(l - 1) * FEAT;
      float* o = bufs[l & 1];
      if (l == 5)
        gemm_conv<4, 1><<<gemm_grid, blk, 0, stream>>>(hin, aggr, he, wt2_l, W, bb,
                                                       o, FEAT, FEAT);
      else
        gemm_conv<4, 0><<<gemm_grid, blk, 0, stream>>>(hin, aggr, he, wt2_l, W, bb,
                                                       o, FEAT, FEAT);
      hin = o;
    }
  }
}